// ProbabilisticAlignmentLoss_34505767256080
// MI455X (gfx1250) — compile-verified
//
#include <hip/hip_runtime.h>
#include <math.h>

// ---------------------------------------------------------------------------
// ProbabilisticAlignmentLoss for MI455X (gfx1250, wave32)
//   x: 512x64 f32, y: 256x64 f32, q: 257x256 f32  ->  scalar f32
//
// Phase 1: dists via f32 WMMA (V_WMMA_F32_16X16X4_F32), dist^2 = xx + yy - 2*x.yT
// Phase 2: sequential reachability scan (1 block, LDS-resident p)
// Phase 3: wavefront DP, Cbar (256KB) resident in CDNA5's 320KB LDS,
//          one wave32 per lattice cell, one barrier per anti-diagonal.
// ---------------------------------------------------------------------------

#define M_DIM 512
#define N_DIM 256
#define D_DIM 64
#define EPSF  1e-6f
#define INFF  1e30f

// workspace offsets (in floats)
#define OFF_DISTS 0            // 131072
#define OFF_XX    131072       // 512
#define OFF_YY    131584       // 256
#define OFF_P     131840       // 257
#define OFF_W     132128       // 65536
#define OFF_PN    197664       // 256
#define OFF_C     197920       // 513*256 = 131328  (row m stored at (m+1)*N; row 0 = INF sentinel for m=-1)

typedef __attribute__((ext_vector_type(2))) float v2f;
typedef __attribute__((ext_vector_type(8))) float v8f;

// --------------------------- Phase 1a: row norms ---------------------------
__global__ void pal_norms_kernel(const float* __restrict__ x,
                                 const float* __restrict__ y,
                                 float* __restrict__ ws) {
    int i = blockIdx.x * blockDim.x + threadIdx.x;
    if (i < M_DIM) {
        const float* r = x + i * D_DIM;
        float s = 0.f;
        #pragma unroll
        for (int k = 0; k < D_DIM; ++k) s += r[k] * r[k];
        ws[OFF_XX + i] = s;
    } else if (i < M_DIM + N_DIM) {
        const float* r = y + (i - M_DIM) * D_DIM;
        float s = 0.f;
        #pragma unroll
        for (int k = 0; k < D_DIM; ++k) s += r[k] * r[k];
        ws[OFF_YY + (i - M_DIM)] = s;
    }
}

// ----------------------- Phase 1b: WMMA distance tiles ---------------------
// One wave32 per 16x16 output tile; 16 x V_WMMA_F32_16X16X4_F32 over K=64.
// A fragment (16x4 f32, 2 VGPRs): lane L -> M = L%16; VGPR0 holds K = (L<16?0:2),
// VGPR1 holds K+1 (ISA 7.12.2).  B (4x16) mirrors with N = L%16.
__global__ void pal_wmma_dists_kernel(const float* __restrict__ x,
                                      const float* __restrict__ y,
                                      float* __restrict__ ws) {
    const int lane = threadIdx.x;      // 0..31
    const int l  = lane & 15;
    const int hi = lane >> 4;          // 0: K+0/K+1, 1: K+2/K+3
    const int tm = blockIdx.x * 16;
    const int tn = blockIdx.y * 16;

    const float* xrow = x + (tm + l) * D_DIM;
    const float* yrow = y + (tn + l) * D_DIM;

    v8f c = {};
    #pragma unroll
    for (int k = 0; k < D_DIM; k += 4) {
        v2f a, b;
        a.x = xrow[k + 2 * hi];
        a.y = xrow[k + 2 * hi + 1];
        b.x = yrow[k + 2 * hi];
        b.y = yrow[k + 2 * hi + 1];
        c = __builtin_amdgcn_wmma_f32_16x16x4_f32(
                /*neg_a=*/false, a, /*neg_b=*/false, b,
                /*c_mod=*/(short)0, c, /*reuse_a=*/false, /*reuse_b=*/false);
    }

    const float* xx = ws + OFF_XX;
    const float  yyn = ws[OFF_YY + tn + l];
    float* dists = ws + OFF_DISTS;
    // C/D layout: VGPR r, lanes 0-15 -> M=r, lanes 16-31 -> M=r+8; N = lane%16
    #pragma unroll
    for (int r = 0; r < 8; ++r) {
        int m = tm + r + 8 * hi;
        float d2 = xx[m] + yyn - 2.0f * c[r];
        dists[m * N_DIM + tn + l] = sqrtf(fmaxf(d2, 0.0f));
    }
}

// ----------------------- Phase 2a: reachability scan -----------------------
__global__ void pal_reach_kernel(const float* __restrict__ q,
                                 float* __restrict__ ws) {
    __shared__ float p[N_DIM + 1];
    const int tid = threadIdx.x;
    if (tid < N_DIM + 1) p[tid] = (tid == 0) ? 1.0f : 0.0f;
    __syncthreads();
    for (int nb = 0; nb < N_DIM; ++nb) {
        float s = p[nb];                 // old value, read by everyone
        __syncthreads();                 // all reads before any write
        if (tid < N_DIM + 1) p[tid] += s * q[tid * N_DIM + nb];
        __syncthreads();                 // writes visible for next step
    }
    if (tid < N_DIM + 1) ws[OFF_P + tid] = p[tid];
}

// ----------------- Phase 2b: p_back -> mixing weights w, pN ----------------
__global__ void pal_pback_kernel(const float* __restrict__ q,
                                 float* __restrict__ ws) {
    const int n  = blockIdx.x;           // 0..256
    const int nb = threadIdx.x;          // 0..255
    const float* p = ws + OFF_P;
    float pb = p[nb] * q[n * N_DIM + nb] / (p[n] + EPSF);
    if (n < N_DIM) {
        // E[n, nb] = (nb < n) && (p_back > 0)
        ws[OFF_W + n * N_DIM + nb] = (nb < n && pb > 0.0f) ? pb : 0.0f;
    } else {
        ws[OFF_PN + nb] = pb;            // terminal row p_back[N]
    }
}

// ------------------------- Phase 3: wavefront DP ---------------------------
// Single workgroup (32 waves). Cbar (N x N f32 = 256KB) lives in LDS.
// Cell (m,n) depends on (m,n-1) and (m-1,n)  =>  anti-diagonal wavefront.
__global__ void __launch_bounds__(1024) pal_dp_kernel(float* __restrict__ ws) {
    extern __shared__ float cbar[];                 // N_DIM * N_DIM floats
    const int tid   = threadIdx.x;
    const int lane  = tid & 31;
    const int wave  = tid >> 5;
    const int nwave = blockDim.x >> 5;

    float* C = ws + OFF_C;                          // row m at (m+1)*N; row 0 = m=-1
    const float* dists = ws + OFF_DISTS;
    const float* w     = ws + OFF_W;

    for (int i = tid; i < N_DIM * N_DIM; i += blockDim.x) cbar[i] = INFF;
    for (int i = tid; i < N_DIM; i += blockDim.x) C[i] = INFF;   // m = -1 sentinel
    __syncthreads();

    for (int d = 0; d < M_DIM + N_DIM - 1; ++d) {
        const int mlo = (d > N_DIM - 1) ? (d - (N_DIM - 1)) : 0;
        const int mhi = (d < M_DIM - 1) ? d : (M_DIM - 1);
        const int ncells = mhi - mlo + 1;

        for (int cell = wave; cell < ncells; cell += nwave) {
            const int m = mlo + cell;
            const int n = d - m;
            const float* Cprev = C + m * N_DIM;        // row m-1
            float*       Ccur  = C + (m + 1) * N_DIM;  // row m
            const float dmn = dists[m * N_DIM + n];

            if (n == 0) {
                if (lane == 0) {
                    float base = (m == 0) ? 0.0f : Cprev[0];
                    Ccur[0] = dmn + base;
                }
                continue;
            }

            float*       cb   = cbar + n * N_DIM;
            const float* wrow = w + n * N_DIM;
            float acc = 0.0f;
            for (int nb = lane; nb < n; nb += 32) {
                float cmin = fminf(Ccur[nb], fminf(Cprev[nb], cb[nb]));
                float cv = dmn + cmin;
                cb[nb] = cv;                           // in-place: read-once then overwrite
                acc += wrow[nb] * cv;
            }
            #pragma unroll
            for (int off = 16; off > 0; off >>= 1)
                acc += __shfl_xor(acc, off, 32);
            if (lane == 0) Ccur[n] = acc;
        }
        __threadfence();      // make global C writes visible across the workgroup
        __syncthreads();      // advance the wavefront
    }
}

// ------------------------- Phase 4: final reduction ------------------------
__global__ void pal_final_kernel(const float* __restrict__ ws,
                                 float* __restrict__ out) {
    __shared__ float red[N_DIM];
    const int tid = threadIdx.x;
    const float* pN    = ws + OFF_PN;
    const float* Clast = ws + OFF_C + M_DIM * N_DIM;   // row M-1
    red[tid] = pN[tid] * Clast[tid];
    __syncthreads();
    for (int s = N_DIM / 2; s > 0; s >>= 1) {
        if (tid < s) red[tid] += red[tid + s];
        __syncthreads();
    }
    if (tid == 0) out[0] = red[0];
}

// ---------------------------------------------------------------------------
extern "C" void kernel_launch(void* const* d_in, const int* in_sizes, int n_in,
                              void* d_out, int out_size, void* d_ws, size_t ws_size,
                              hipStream_t stream) {
    (void)in_sizes; (void)n_in; (void)out_size; (void)ws_size;
    const float* x = (const float*)d_in[0];
    const float* y = (const float*)d_in[1];
    const float* q = (const float*)d_in[2];
    float* out = (float*)d_out;
    float* ws  = (float*)d_ws;

    // Phase 1: distances (WMMA)
    pal_norms_kernel<<<(M_DIM + N_DIM + 255) / 256, 256, 0, stream>>>(x, y, ws);
    pal_wmma_dists_kernel<<<dim3(M_DIM / 16, N_DIM / 16), 32, 0, stream>>>(x, y, ws);

    // Phase 2: reachability + mixing weights
    pal_reach_kernel<<<1, 288, 0, stream>>>(q, ws);
    pal_pback_kernel<<<N_DIM + 1, N_DIM, 0, stream>>>(q, ws);

    // Phase 3: wavefront DP (single workgroup, 256KB LDS Cbar)
    pal_dp_kernel<<<1, 1024, N_DIM * N_DIM * sizeof(float), stream>>>(ws);

    // Phase 4: terminal expectation
    pal_final_kernel<<<1, N_DIM, 0, stream>>>(ws, out);
}